// GCN_83966610637081
// MI455X (gfx1250) — compile-verified
//
#include <hip/hip_runtime.h>
#include <hip/hip_bf16.h>

// ---------------------------------------------------------------------------
// GCN 2-layer forward for MI455X (gfx1250, wave32).
//   layer: h = x@W ; agg[d] += norm(s,d)*h[s] over edges+selfloops ; +bias
//   leaky_relu between layers.
// GEMMs use V_WMMA_F32_16X16X4_F32 (full-f32 matrix core, matches reference
// precision). Aggregation uses float4 gathers + native f32 global atomics;
// h / accumulator tiles are L2-resident (192MB L2 >> 12.8MB per buffer).
// ---------------------------------------------------------------------------

typedef float v2f __attribute__((ext_vector_type(2)));
typedef float v8f __attribute__((ext_vector_type(8)));

#define IN_F 128
#define HID_F 64
#define OUT_F 64

// ---------------- degree / normalization ----------------

__global__ void deg_init_kernel(float* deg, int n) {
    int i = blockIdx.x * blockDim.x + threadIdx.x;
    if (i < n) deg[i] = 1.0f;  // self-loop contribution
}

__global__ void deg_count_kernel(const int* __restrict__ dst, float* deg, int e) {
    int i = blockIdx.x * blockDim.x + threadIdx.x;
    if (i < e) {
        __hip_atomic_fetch_add(&deg[dst[i]], 1.0f,
                               __ATOMIC_RELAXED, __HIP_MEMORY_SCOPE_AGENT);
    }
}

__global__ void dinv_kernel(const float* __restrict__ deg, float* dinv, int n) {
    int i = blockIdx.x * blockDim.x + threadIdx.x;
    if (i < n) {
        float d = deg[i];
        dinv[i] = (d > 0.0f) ? rsqrtf(d) : 0.0f;
    }
}

// ---------------- WMMA GEMM: H[N x 64] = A[N x K] @ W[K x 64] ----------------
// Block = 128 threads (4 waves). Block computes 16 rows x 64 cols.
// Wave w computes the 16x16 tile at columns [16w, 16w+16).
// K-loop steps by 4 using v_wmma_f32_16x16x4_f32.
//
// Per-lane VGPR layouts (CDNA5 ISA §7.12.2):
//   A (16x4, M x K):  lanes 0-15: a.x=K0,a.y=K1 ; lanes 16-31: a.x=K2,a.y=K3 ; M=lane&15
//   B (4x16, K x N):  lanes 0-15: b.x=row K0, b.y=row K1 ; lanes 16-31: K2/K3 ; N=lane&15
//   C/D (16x16):      vgpr i -> M = i + 8*(lane>>4), N = lane&15
__global__ void __launch_bounds__(128)
gcn_gemm_wmma_kernel(const float* __restrict__ A, const float* __restrict__ W,
                     float* __restrict__ H, int K, int nrows) {
    extern __shared__ float smem[];
    const int lda = K + 1;                 // pad: stride K=128 would 16-way conflict
    float* As = smem;                      // 16 * lda
    float* Ws = smem + 16 * lda;           // K * 64

    const int tid = threadIdx.x;
    const int rowBase = blockIdx.x * 16;

    // Stage full weight matrix (<= 32KB) — coalesced.
    for (int i = tid; i < K * 64; i += 128) Ws[i] = W[i];
    // Stage 16 x K activation tile — coalesced along K.
    for (int i = tid; i < 16 * K; i += 128) {
        int r = i / K, k = i - r * K;
        int row = rowBase + r;
        if (row >= nrows) row = nrows - 1;   // clamp (no EXEC divergence)
        As[r * lda + k] = A[(size_t)row * K + k];
    }
    __syncthreads();

    const int lane = tid & 31;
    const int wave = tid >> 5;
    const int m    = lane & 15;
    const int half = lane >> 4;
    const int n0   = wave * 16;

    v8f c = {0.f, 0.f, 0.f, 0.f, 0.f, 0.f, 0.f, 0.f};
    for (int kk = 0; kk < K; kk += 4) {
        const int ka = kk + 2 * half;
        v2f a, b;
        a.x = As[m * lda + ka];
        a.y = As[m * lda + ka + 1];
        b.x = Ws[ka * 64 + n0 + m];
        b.y = Ws[(ka + 1) * 64 + n0 + m];
        // D = A x B + C  (full fp32; EXEC is all-1s here)
        c = __builtin_amdgcn_wmma_f32_16x16x4_f32(
                /*neg_a=*/false, a, /*neg_b=*/false, b,
                /*c_mod=*/(short)0, c, /*reuse_a=*/false, /*reuse_b=*/false);
    }

#pragma unroll
    for (int i = 0; i < 8; ++i) {
        int row = rowBase + i + 8 * half;
        if (row < nrows) H[(size_t)row * 64 + n0 + m] = c[i];
    }
}

// ---------------- bias broadcast init ----------------

__global__ void bias_init_kernel(float* out, const float* __restrict__ b, long total) {
    long i = (long)blockIdx.x * blockDim.x + threadIdx.x;
    if (i < total) out[i] = b[i & 63];
}

// ---------------- edge aggregation (gather * norm -> atomic scatter-add) ----
// Work item granularity: 16 threads per edge, each owning one float4 of the
// 64-float feature row. Indices >= E are self-loops (s = d = idx - E).

__global__ void aggregate_kernel(const float* __restrict__ h,
                                 const int* __restrict__ src,
                                 const int* __restrict__ dst,
                                 const float* __restrict__ dinv,
                                 float* out, int e, int total) {
    long gid = (long)blockIdx.x * blockDim.x + threadIdx.x;
    int  idx = (int)(gid >> 4);
    if (idx >= total) return;
    int q = (int)(gid & 15);

    int s, d;
    if (idx < e) { s = src[idx]; d = dst[idx]; }
    else         { s = d = idx - e; }

    float w = dinv[s] * dinv[d];

    const float4 v = ((const float4*)h)[(size_t)s * 16 + q];
    float* o = out + (size_t)d * 64 + q * 4;
    __hip_atomic_fetch_add(o + 0, v.x * w, __ATOMIC_RELAXED, __HIP_MEMORY_SCOPE_AGENT);
    __hip_atomic_fetch_add(o + 1, v.y * w, __ATOMIC_RELAXED, __HIP_MEMORY_SCOPE_AGENT);
    __hip_atomic_fetch_add(o + 2, v.z * w, __ATOMIC_RELAXED, __HIP_MEMORY_SCOPE_AGENT);
    __hip_atomic_fetch_add(o + 3, v.w * w, __ATOMIC_RELAXED, __HIP_MEMORY_SCOPE_AGENT);
}

// ---------------- leaky relu (in place) ----------------

__global__ void leaky_relu_kernel(float* x, long total) {
    long i = (long)blockIdx.x * blockDim.x + threadIdx.x;
    if (i < total) {
        float v = x[i];
        x[i] = (v > 0.0f) ? v : 0.01f * v;
    }
}

// ---------------------------------------------------------------------------

extern "C" void kernel_launch(void* const* d_in, const int* in_sizes, int n_in,
                              void* d_out, int out_size, void* d_ws, size_t ws_size,
                              hipStream_t stream) {
    const float* x   = (const float*)d_in[0];
    const int*   ei  = (const int*)d_in[1];
    const float* W1  = (const float*)d_in[2];
    const float* b1  = (const float*)d_in[3];
    const float* W2  = (const float*)d_in[4];
    const float* b2  = (const float*)d_in[5];
    float*       out = (float*)d_out;

    const int N = in_sizes[0] / IN_F;   // 50000
    const int E = in_sizes[1] / 2;      // 800000
    const int T = E + N;                // edges + self-loops

    const int* src = ei;
    const int* dst = ei + E;

    // workspace carve-up (floats)
    float* deg  = (float*)d_ws;
    float* dinv = deg  + N;
    float* h1   = dinv + N;              // N x 64
    float* a1   = h1   + (size_t)N * 64; // N x 64 (aggregated + relu)
    float* h2   = a1   + (size_t)N * 64; // N x 64

    const int B = 256;

    // 1) degrees and symmetric normalization
    deg_init_kernel<<<(N + B - 1) / B, B, 0, stream>>>(deg, N);
    deg_count_kernel<<<(E + B - 1) / B, B, 0, stream>>>(dst, deg, E);
    dinv_kernel<<<(N + B - 1) / B, B, 0, stream>>>(deg, dinv, N);

    const int gemmGrid = (N + 15) / 16;  // 3125 blocks, 4 waves each

    // 2) layer 1 GEMM: h1 = x @ W1   (K = 128)
    {
        size_t smem = (size_t)(16 * (IN_F + 1) + IN_F * 64) * sizeof(float);
        gcn_gemm_wmma_kernel<<<gemmGrid, 128, smem, stream>>>(x, W1, h1, IN_F, N);
    }

    // 3) layer 1 aggregation: a1 = b1 + scatter(norm * h1[src])
    {
        long tot = (long)N * 64;
        bias_init_kernel<<<(tot + B - 1) / B, B, 0, stream>>>(a1, b1, tot);
        long work = (long)T * 16;
        aggregate_kernel<<<(work + B - 1) / B, B, 0, stream>>>(h1, src, dst, dinv, a1, E, T);
        leaky_relu_kernel<<<(tot + B - 1) / B, B, 0, stream>>>(a1, tot);
    }

    // 4) layer 2 GEMM: h2 = a1 @ W2  (K = 64)
    {
        size_t smem = (size_t)(16 * (HID_F + 1) + HID_F * 64) * sizeof(float);
        gcn_gemm_wmma_kernel<<<gemmGrid, 128, smem, stream>>>(a1, W2, h2, HID_F, N);
    }

    // 5) layer 2 aggregation straight into d_out
    {
        long tot = (long)N * 64;
        bias_init_kernel<<<(tot + B - 1) / B, B, 0, stream>>>(out, b2, tot);
        long work = (long)T * 16;
        aggregate_kernel<<<(work + B - 1) / B, B, 0, stream>>>(h2, src, dst, dinv, out, E, T);
    }
}